// GATEncoder_2491081031686
// MI455X (gfx1250) — compile-verified
//
#include <hip/hip_runtime.h>
#include <cstdint>

typedef __attribute__((ext_vector_type(2))) float v2f;
typedef __attribute__((ext_vector_type(8))) float v8f;

// ---------- helpers ----------
__device__ __forceinline__ int enc_ord(float f) {
  int i = __float_as_int(f);
  return i >= 0 ? i : (i ^ 0x7fffffff);
}
__device__ __forceinline__ float dec_ord(int i) {
  int j = i >= 0 ? i : (i ^ 0x7fffffff);
  return __int_as_float(j);
}

// ---------- generic fill ----------
__global__ void __launch_bounds__(256)
fill_u32_kernel(uint32_t* __restrict__ p, uint32_t v, long long n) {
  long long i = (long long)blockIdx.x * blockDim.x + threadIdx.x;
  if (i < n) p[i] = v;
}

// ---------- fp32 WMMA GEMM: C[M,N] = A[M,K] * B[K,N], row-major ----------
// Block (8 waves) computes a 128x64 tile. B panel (K x 64) is staged in LDS
// once per block (stride 65 to avoid lane-half bank conflicts); each wave
// computes 16x64 with 4 accumulators so one A fragment feeds 4 WMMAs.
#define BS_STRIDE 65

__global__ void __launch_bounds__(256)
gemm_wmma_f32(const float* __restrict__ A, const float* __restrict__ B,
              float* __restrict__ C, int M, int N, int K) {
  __shared__ float Bs[128 * BS_STRIDE];  // up to 128-deep K chunk, 64 cols

  const int lane = threadIdx.x & 31;
  const int wave = threadIdx.x >> 5;
  const int half = lane >> 4;            // selects K pair {0,1} vs {2,3}
  const int r    = lane & 15;            // row (A) / col (B,C) within tile

  const int tilesM  = (M + 15) >> 4;
  const int tilesN  = (N + 15) >> 4;
  const int groupsN = (tilesN + 3) >> 2;

  const int bm = blockIdx.x / groupsN;
  const int bn = blockIdx.x % groupsN;

  int tm = bm * 8 + wave;
  if (tm >= tilesM) tm = tilesM - 1;     // clamp: duplicate work, identical stores
  const int cn0 = bn * 64;               // first global column of the panel

  int arow = tm * 16 + r;
  if (arow >= M) arow = M - 1;
  const float* __restrict__ Ap = A + (size_t)arow * K;

  v8f acc0 = {}, acc1 = {}, acc2 = {}, acc3 = {};

  for (int k0 = 0; k0 < K; k0 += 128) {
    const int kc = (K - k0) < 128 ? (K - k0) : 128;   // chunk depth (mult of 4)

    // cooperative coalesced load of B[k0:k0+kc, cn0:cn0+64] into LDS
    for (int i = threadIdx.x; i < kc * 64; i += 256) {
      const int kk = i >> 6, cc = i & 63;
      const int gcol = cn0 + cc;
      Bs[kk * BS_STRIDE + cc] =
          (gcol < N) ? B[(size_t)(k0 + kk) * N + gcol] : 0.f;
    }
    __syncthreads();

    for (int k = 0; k < kc; k += 4) {
      const int ka = k + 2 * half;                 // this half-wave's K offset
      const v2f a = *(const v2f*)(Ap + k0 + ka);   // one 8B load, feeds 4 WMMAs
      const float* __restrict__ brow0 = Bs + (size_t)ka * BS_STRIDE;
      const float* __restrict__ brow1 = brow0 + BS_STRIDE;
      v2f b;
      b.x = brow0[r];       b.y = brow1[r];
      acc0 = __builtin_amdgcn_wmma_f32_16x16x4_f32(false, a, false, b,
                                                   (short)0, acc0, false, false);
      b.x = brow0[16 + r];  b.y = brow1[16 + r];
      acc1 = __builtin_amdgcn_wmma_f32_16x16x4_f32(false, a, false, b,
                                                   (short)0, acc1, false, false);
      b.x = brow0[32 + r];  b.y = brow1[32 + r];
      acc2 = __builtin_amdgcn_wmma_f32_16x16x4_f32(false, a, false, b,
                                                   (short)0, acc2, false, false);
      b.x = brow0[48 + r];  b.y = brow1[48 + r];
      acc3 = __builtin_amdgcn_wmma_f32_16x16x4_f32(false, a, false, b,
                                                   (short)0, acc3, false, false);
    }
    __syncthreads();
  }

  // store: C/D layout -> lane half selects M = vgpr + 8*half, lane r = column
  const int crow0 = tm * 16 + 8 * half;
#pragma unroll
  for (int i = 0; i < 8; ++i) {
    const int row = crow0 + i;
    if (row >= M) continue;
    float* __restrict__ Cp = C + (size_t)row * N + cn0 + r;
    if (cn0 + r      < N) Cp[0]  = acc0[i];
    if (cn0 + 16 + r < N) Cp[16] = acc1[i];
    if (cn0 + 32 + r < N) Cp[32] = acc2[i];
    if (cn0 + 48 + r < N) Cp[48] = acc3[i];
  }
}

// ---------- per-(node,head) attention logits: one wave32 each ----------
__global__ void __launch_bounds__(256)
alpha_kernel(const float* __restrict__ h, const float* __restrict__ a_src,
             const float* __restrict__ a_dst, float* __restrict__ as,
             float* __restrict__ ad, int NH, int H, int C) {
  const int lane = threadIdx.x & 31;
  const int wid  = blockIdx.x * 8 + (threadIdx.x >> 5);
  if (wid >= NH) return;
  const int hh = wid % H;
  const float* __restrict__ hp  = h + (size_t)wid * C;
  const float* __restrict__ asp = a_src + (size_t)hh * C;
  const float* __restrict__ adp = a_dst + (size_t)hh * C;
  float ss = 0.f, sd = 0.f;
  for (int c = lane; c < C; c += 32) {
    const float v = hp[c];
    ss += v * asp[c];
    sd += v * adp[c];
  }
  for (int off = 16; off; off >>= 1) {
    ss += __shfl_xor(ss, off, 32);
    sd += __shfl_xor(sd, off, 32);
  }
  if (lane == 0) { as[wid] = ss; ad[wid] = sd; }
}

// ---------- edge pass A: leaky_relu logits + segment max (ordered-int) ----------
__global__ void __launch_bounds__(256)
edge_max_kernel(const int* __restrict__ ei, int E, int N, int H,
                const float* __restrict__ as, const float* __restrict__ ad,
                float* __restrict__ ebuf, int* __restrict__ mmax) {
  const int e = blockIdx.x * blockDim.x + threadIdx.x;
  const int ET = E + N;
  if (e >= ET) return;
  int s, d;
  if (e < E) { s = ei[e]; d = ei[E + e]; } else { s = d = e - E; }
  for (int hh = 0; hh < H; ++hh) {
    float v = as[(size_t)s * H + hh] + ad[(size_t)d * H + hh];
    v = v > 0.f ? v : 0.2f * v;                    // leaky_relu(0.2)
    ebuf[(size_t)e * H + hh] = v;
    atomicMax(&mmax[(size_t)d * H + hh], enc_ord(v));
  }
}

// ---------- edge pass B: p=exp(e-m); denom += p; agg[dst] += p*h[src] ----------
// One wave32 per (edge,head); 4 channels per lane (C==128), float4 gathers,
// non-returning global_atomic_add_f32 scatters (all resident in L2).
__global__ void __launch_bounds__(256)
edge_agg_kernel(const int* __restrict__ ei, int E, int N, int H, int C,
                const float* __restrict__ ebuf, const int* __restrict__ mmax,
                const float* __restrict__ hfeat, float* __restrict__ denom,
                float* __restrict__ agg) {
  const int lane = threadIdx.x & 31;
  const long long wid = (long long)blockIdx.x * 8 + (threadIdx.x >> 5);
  const long long total = (long long)(E + N) * H;
  if (wid >= total) return;
  const int e  = (int)(wid / H);
  const int hh = (int)(wid % H);
  int s, d;
  if (e < E) { s = ei[e]; d = ei[E + e]; } else { s = d = e - E; }
  const float m = dec_ord(mmax[(size_t)d * H + hh]);
  const float p = __expf(ebuf[wid] - m);
  if (lane == 0) atomicAdd(&denom[(size_t)d * H + hh], p);
  const float4 hv =
      *(const float4*)(hfeat + ((size_t)s * H + hh) * C + lane * 4);
  float* ap = agg + ((size_t)d * H + hh) * C + lane * 4;
  atomicAdd(ap + 0, p * hv.x);
  atomicAdd(ap + 1, p * hv.y);
  atomicAdd(ap + 2, p * hv.z);
  atomicAdd(ap + 3, p * hv.w);
}

// ---------- finalize: normalize, mean over heads, +bias, ELU ----------
__global__ void __launch_bounds__(256)
finalize_kernel(const float* __restrict__ agg, const float* __restrict__ denom,
                const float* __restrict__ bias, float* __restrict__ out,
                int N, int H, int C) {
  const long long i = (long long)blockIdx.x * blockDim.x + threadIdx.x;
  if (i >= (long long)N * C) return;
  const int n = (int)(i / C);
  const int c = (int)(i % C);
  float acc = 0.f;
  for (int hh = 0; hh < H; ++hh)
    acc += agg[((size_t)n * H + hh) * C + c] / denom[(size_t)n * H + hh];
  acc = acc / (float)H + bias[c];
  out[i] = acc > 0.f ? acc : (__expf(acc) - 1.f);   // ELU(alpha=1)
}

// ---------- host ----------
static inline int iceil(long long a, long long b) { return (int)((a + b - 1) / b); }

extern "C" void kernel_launch(void* const* d_in, const int* in_sizes, int n_in,
                              void* d_out, int out_size, void* d_ws, size_t ws_size,
                              hipStream_t stream) {
  const float* x      = (const float*)d_in[0];
  const float* W1     = (const float*)d_in[1];
  const float* a_src1 = (const float*)d_in[2];
  const float* a_dst1 = (const float*)d_in[3];
  const float* b1     = (const float*)d_in[4];
  const float* W2     = (const float*)d_in[5];
  const float* a_src2 = (const float*)d_in[6];
  const float* a_dst2 = (const float*)d_in[7];
  const float* b2     = (const float*)d_in[8];
  const int*   ei     = (const int*)d_in[9];

  const int F = in_sizes[4];            // 128
  const int N = in_sizes[0] / F;        // 20000
  const int H = in_sizes[2] / F;        // 4
  const int E = in_sizes[9] / 2;        // 640000
  const int C = F;
  const int ET = E + N;                 // edges + self loops

  // workspace layout (floats), with reuse across layers
  float* ws = (float*)d_ws;
  size_t o = 0;
  float* h1   = ws + o; o += (size_t)N * H * C;     // layer1 features
  float* agg1 = ws + o; o += (size_t)N * H * C;     // layer1 accum; reused as h2
  float* out1 = ws + o; o += (size_t)N * C;         // layer1 output
  float* e1   = ws + o; o += (size_t)ET * H;        // logits; reused layer2
  int*   mx   = (int*)(ws + o); o += (size_t)N * H; // seg-max; reused layer2
  float* den  = ws + o; o += (size_t)N * H;         // softmax denom; reused
  float* as   = ws + o; o += (size_t)N * H;         // alpha_src; reused
  float* ad   = ws + o; o += (size_t)N * H;         // alpha_dst; reused
  float* h2   = agg1;
  float* outF = (float*)d_out;                      // layer2 accum in-place

  const uint32_t INT_MIN_U = 0x80000000u;

  // ===== Layer 1 (H heads) =====
  fill_u32_kernel<<<iceil((long long)N * H, 256), 256, 0, stream>>>(
      (uint32_t*)mx, INT_MIN_U, (long long)N * H);
  fill_u32_kernel<<<iceil((long long)N * H, 256), 256, 0, stream>>>(
      (uint32_t*)den, 0u, (long long)N * H);
  fill_u32_kernel<<<iceil((long long)N * H * C, 256), 256, 0, stream>>>(
      (uint32_t*)agg1, 0u, (long long)N * H * C);

  {
    const int N1 = H * C;
    const int blocks = iceil((N + 15) / 16, 8) * iceil((N1 + 15) / 16, 4);
    gemm_wmma_f32<<<blocks, 256, 0, stream>>>(x, W1, h1, N, N1, F);
  }
  alpha_kernel<<<iceil((long long)N * H, 8), 256, 0, stream>>>(
      h1, a_src1, a_dst1, as, ad, N * H, H, C);
  edge_max_kernel<<<iceil(ET, 256), 256, 0, stream>>>(
      ei, E, N, H, as, ad, e1, mx);
  edge_agg_kernel<<<iceil((long long)ET * H, 8), 256, 0, stream>>>(
      ei, E, N, H, C, e1, mx, h1, den, agg1);
  finalize_kernel<<<iceil((long long)N * C, 256), 256, 0, stream>>>(
      agg1, den, b1, out1, N, H, C);

  // ===== Layer 2 (1 head) =====
  fill_u32_kernel<<<iceil((long long)N, 256), 256, 0, stream>>>(
      (uint32_t*)mx, INT_MIN_U, (long long)N);
  fill_u32_kernel<<<iceil((long long)N, 256), 256, 0, stream>>>(
      (uint32_t*)den, 0u, (long long)N);
  fill_u32_kernel<<<iceil((long long)N * C, 256), 256, 0, stream>>>(
      (uint32_t*)outF, 0u, (long long)N * C);

  {
    const int blocks = iceil((N + 15) / 16, 8) * iceil((C + 15) / 16, 4);
    gemm_wmma_f32<<<blocks, 256, 0, stream>>>(out1, W2, h2, N, C, C);
  }
  alpha_kernel<<<iceil((long long)N, 8), 256, 0, stream>>>(
      h2, a_src2, a_dst2, as, ad, N, 1, C);
  edge_max_kernel<<<iceil(ET, 256), 256, 0, stream>>>(
      ei, E, N, 1, as, ad, e1, mx);
  edge_agg_kernel<<<iceil((long long)ET, 8), 256, 0, stream>>>(
      ei, E, N, 1, C, e1, mx, h2, den, outF);
  finalize_kernel<<<iceil((long long)N * C, 256), 256, 0, stream>>>(
      outF, den, b2, outF, N, 1, C);
}